// MoNet_23218593202711
// MI455X (gfx1250) — compile-verified
//
#include <hip/hip_runtime.h>
#include <hip/hip_bf16.h>

// ---------------------------------------------------------------------------
// MoNet (GMM graph conv), 2 layers.
//   N=50000 nodes, E=800000 edges, IN_F=128, HID=64, OUT_F=64, DIM=2, K=3
// Node GEMMs via V_WMMA_F32_16X16X4_F32; edge gather + global f32 atomics.
// ---------------------------------------------------------------------------

typedef __attribute__((ext_vector_type(2))) float v2f;
typedef __attribute__((ext_vector_type(8))) float v8f;

// C[N,192] = A[N,FIN] @ B[FIN,192]
// One wave32 per 16-row block; 12 WMMA accumulator tiles (full 192 cols)
// so each A element is loaded exactly once per block.
//
// A 16x4 f32 layout (ISA 7.12.2): lanes 0-15 -> M=lane, v0=K0,v1=K1;
//                                 lanes 16-31 -> M=lane-16, v0=K2,v1=K3.
// B 4x16: row striped across lanes within a VGPR (v0: K0|K2, v1: K1|K3).
// C/D 16x16: vgpr r -> (M=r, N=lane) lanes 0-15; (M=r+8, N=lane-16) hi half.
template <int FIN>
__global__ __launch_bounds__(32) void gemm_wmma_f32(
    const float* __restrict__ A, const float* __restrict__ B,
    float* __restrict__ C, int N) {
  const int lane = threadIdx.x & 31;
  const int nIdx = lane & 15;   // M-row of A strip / N-col of B,C tiles
  const int half = lane >> 4;   // selects K pair {0,1} vs {2,3}
  const int m0   = blockIdx.x * 16;

  v8f acc[12];
#pragma unroll
  for (int t = 0; t < 12; ++t) acc[t] = (v8f)(0.0f);

  int arow = m0 + nIdx;
  if (arow > N - 1) arow = N - 1;           // clamp (N is multiple of 16 here)
  const float* Ap = A + (size_t)arow * FIN + 2 * half;

  for (int k0 = 0; k0 < FIN; k0 += 4) {
    v2f a;
    a.x = Ap[k0];                           // K = k0 + 2*half
    a.y = Ap[k0 + 1];                       // K = k0 + 2*half + 1
    const float* Bp = B + (size_t)(k0 + 2 * half) * 192 + nIdx;
#pragma unroll
    for (int t = 0; t < 12; ++t) {
      v2f b;
      b.x = Bp[t * 16];                     // row k0+2*half
      b.y = Bp[t * 16 + 192];               // row k0+2*half+1
      acc[t] = __builtin_amdgcn_wmma_f32_16x16x4_f32(
          /*neg_a=*/false, a, /*neg_b=*/false, b,
          /*c_mod=*/(short)0, acc[t], /*reuse_a=*/false, /*reuse_b=*/false);
    }
  }

#pragma unroll
  for (int t = 0; t < 12; ++t) {
#pragma unroll
    for (int r = 0; r < 8; ++r) {
      int gm = m0 + r + 8 * half;
      if (gm < N) C[(size_t)gm * 192 + t * 16 + nIdx] = acc[t][r];
    }
  }
}

// Seed the aggregation buffer with the per-feature bias (F == 64).
__global__ __launch_bounds__(256) void bias_bcast(
    float* __restrict__ out, const float* __restrict__ bias, int total) {
  int i = blockIdx.x * blockDim.x + threadIdx.x;
  if (i < total) out[i] = bias[i & 63];
}

// One wave32 per edge; each lane handles features f=lane and f=lane+32.
// hp row layout: [k0: f0..63 | k1: f0..63 | k2: f0..63]  (768B, coalesced).
__global__ __launch_bounds__(256) void gmm_edge(
    const float* __restrict__ pseudo,                       // [E,2]
    const int* __restrict__ src, const int* __restrict__ dst,
    const float* __restrict__ pw, const float* __restrict__ pb,   // [2,2],[2]
    const float* __restrict__ mu, const float* __restrict__ isg,  // [3,2]
    const float* __restrict__ hp,                           // [N, 3*64]
    float* __restrict__ agg,                                // [N, 64]
    int nE) {
  int e = (int)((blockIdx.x * blockDim.x + threadIdx.x) >> 5);
  if (e >= nE) return;
  const int lane = threadIdx.x & 31;

  // u = tanh(pseudo @ pw + pb)  (wave-uniform scalar work)
  float px = pseudo[2 * e], py = pseudo[2 * e + 1];
  float u0 = tanhf(fmaf(px, pw[0], fmaf(py, pw[2], pb[0])));
  float u1 = tanhf(fmaf(px, pw[1], fmaf(py, pw[3], pb[1])));

  // Gaussian mixture kernel weights w[k]
  float w[3];
#pragma unroll
  for (int k = 0; k < 3; ++k) {
    float d0 = (u0 - mu[2 * k]) * isg[2 * k];
    float d1 = (u1 - mu[2 * k + 1]) * isg[2 * k + 1];
    w[k] = __expf(-0.5f * (d0 * d0 + d1 * d1));
  }

  const float* h = hp + (size_t)src[e] * 192;
  float m0 = w[0] * h[lane]       + w[1] * h[64 + lane]  + w[2] * h[128 + lane];
  float m1 = w[0] * h[32 + lane]  + w[1] * h[96 + lane]  + w[2] * h[160 + lane];

  float* o = agg + (size_t)dst[e] * 64;
  atomicAdd(o + lane, m0);
  atomicAdd(o + lane + 32, m1);
}

extern "C" void kernel_launch(void* const* d_in, const int* in_sizes, int n_in,
                              void* d_out, int out_size, void* d_ws, size_t ws_size,
                              hipStream_t stream) {
  const float* feat   = (const float*)d_in[0];
  const float* pseudo = (const float*)d_in[1];
  const int*   src    = (const int*)d_in[2];
  const int*   dst    = (const int*)d_in[3];
  const float* fc0    = (const float*)d_in[4];
  const float* mu0    = (const float*)d_in[5];
  const float* isg0   = (const float*)d_in[6];
  const float* bias0  = (const float*)d_in[7];
  const float* pw0    = (const float*)d_in[8];
  const float* pb0    = (const float*)d_in[9];
  const float* fc1    = (const float*)d_in[10];
  const float* mu1    = (const float*)d_in[11];
  const float* isg1   = (const float*)d_in[12];
  const float* bias1  = (const float*)d_in[13];
  const float* pw1    = (const float*)d_in[14];
  const float* pb1    = (const float*)d_in[15];
  float* out = (float*)d_out;

  const int N = in_sizes[0] / 128;   // 50000
  const int E = in_sizes[2];         // 800000

  // Workspace: hp [N,192] then h [N,64]  (~51.2 MB total)
  float* hp = (float*)d_ws;
  float* h  = hp + (size_t)N * 192;

  const int mBlocks = (N + 15) / 16;                   // 3125
  const int edgeBlocks = (int)(((size_t)E * 32 + 255) / 256);
  const int nodeElems = N * 64;
  const int nodeBlocks = (nodeElems + 255) / 256;

  // ---- Layer 0: in(128) -> hidden(64) ----
  gemm_wmma_f32<128><<<mBlocks, 32, 0, stream>>>(feat, fc0, hp, N);
  bias_bcast<<<nodeBlocks, 256, 0, stream>>>(h, bias0, nodeElems);
  gmm_edge<<<edgeBlocks, 256, 0, stream>>>(pseudo, src, dst, pw0, pb0,
                                           mu0, isg0, hp, h, E);

  // ---- Layer 1: hidden(64) -> out(64) ----
  gemm_wmma_f32<64><<<mBlocks, 32, 0, stream>>>(h, fc1, hp, N);
  bias_bcast<<<nodeBlocks, 256, 0, stream>>>(out, bias1, nodeElems);
  gmm_edge<<<edgeBlocks, 256, 0, stream>>>(pseudo, src, dst, pw1, pb1,
                                           mu1, isg1, hp, out, E);
}